// HeteroGNN_5729486373617
// MI455X (gfx1250) — compile-verified
//
#include <hip/hip_runtime.h>
#include <hip/hip_bf16.h>

// ---------------------------------------------------------------------------
// Problem constants (match reference)
// ---------------------------------------------------------------------------
static constexpr int N_INT = 100000;
static constexpr int N_EXT = 50000;
static constexpr int D     = 64;
static constexpr int H     = 64;
static constexpr int C     = 2;
static constexpr int E_TXN = 1600000;
static constexpr int E_WD  = 800000;
static constexpr int E_DEP = 800000;
static constexpr float SLOPE = 0.2f;

typedef __attribute__((ext_vector_type(2))) float v2f;
typedef __attribute__((ext_vector_type(8))) float v8f;

// ---------------------------------------------------------------------------
// Monotone float<->uint encoding so atomicMax(uint) == float max
// ---------------------------------------------------------------------------
__device__ __forceinline__ unsigned fenc(float f) {
    unsigned u = __float_as_uint(f);
    return (u & 0x80000000u) ? ~u : (u | 0x80000000u);
}
__device__ __forceinline__ float fdec(unsigned u) {
    return __uint_as_float((u & 0x80000000u) ? (u & 0x7FFFFFFFu) : ~u);
}

// ---------------------------------------------------------------------------
// Zero-fill helpers (graph-capture safe; no hipMemset)
// ---------------------------------------------------------------------------
__global__ void zero_f32_kernel(float* __restrict__ p, int n) {
    int i = blockIdx.x * blockDim.x + threadIdx.x;
    if (i < n) p[i] = 0.0f;
}
__global__ void zero_u32_kernel(unsigned* __restrict__ p, int n) {
    int i = blockIdx.x * blockDim.x + threadIdx.x;
    if (i < n) p[i] = 0u;   // decodes below enc(-FLT_MAX): acts as -inf
}

// ---------------------------------------------------------------------------
// wv[d] = sum_h W[d,h] * att[h]   (W is [64,64] row-major, att [64])
// ---------------------------------------------------------------------------
__global__ void matvec64_kernel(const float* __restrict__ W,
                                const float* __restrict__ att,
                                float* __restrict__ wv) {
    int d = threadIdx.x;            // launched with 64 threads
    float s = 0.0f;
    #pragma unroll
    for (int h = 0; h < H; ++h) s += W[d * H + h] * att[h];
    wv[d] = s;
}

// ---------------------------------------------------------------------------
// a[n] = dot(X[n,:], wv)  -- folded attention scalar
// ---------------------------------------------------------------------------
__global__ void node_dot_kernel(const float* __restrict__ X,
                                const float* __restrict__ wv,
                                float* __restrict__ out, int N) {
    int n = blockIdx.x * blockDim.x + threadIdx.x;
    if (n >= N) return;
    const float* x = X + (size_t)n * D;
    float s = 0.0f;
    #pragma unroll
    for (int d = 0; d < D; d += 4) {
        float4 v = *(const float4*)(x + d);
        s += v.x * wv[d] + v.y * wv[d + 1] + v.z * wv[d + 2] + v.w * wv[d + 3];
    }
    out[n] = s;
}

// ---------------------------------------------------------------------------
// WMMA GEMM:  Y[N,64] = X[N,64] @ W[64,64]   (fp32, V_WMMA_F32_16X16X4_F32)
// One wave computes a 16-row x 64-col strip: 16 K-chunks x 4 N-tiles.
// A layout (16x4 f32): lanes 0-15 -> M=0..15; lane-half selects K pair.
// C/D layout: VGPR i -> rows (base+i, base+i+8) split by lane half.
// ---------------------------------------------------------------------------
__global__ void gemm_wmma_kernel(const float* __restrict__ X,
                                 const float* __restrict__ W,
                                 float* __restrict__ Y, int N) {
    int wave = (blockIdx.x * blockDim.x + threadIdx.x) >> 5;
    int lane = threadIdx.x & 31;
    int row0 = wave * 16;
    if (row0 >= N) return;          // wave-uniform: EXEC stays all-ones

    int r    = lane & 15;           // M (for A) / N (for B) index
    int koff = (lane >> 4) * 2;     // which K pair this lane half holds
    int arow = row0 + r;
    if (arow >= N) arow = N - 1;    // tail clamp (stores are guarded)

    const float* xrow = X + (size_t)arow * D;

    v8f acc[4] = {};
    #pragma unroll
    for (int k = 0; k < D; k += 4) {
        v2f a;
        a.x = xrow[k + koff];
        a.y = xrow[k + koff + 1];
        #pragma unroll
        for (int t = 0; t < 4; ++t) {
            int col = t * 16 + r;
            v2f b;
            b.x = W[(k + koff) * H + col];
            b.y = W[(k + koff + 1) * H + col];
            acc[t] = __builtin_amdgcn_wmma_f32_16x16x4_f32(
                false, a, false, b, (short)0, acc[t], false, false);
        }
    }

    int mo = (lane >> 4) * 8;       // lanes 16-31 hold rows i+8
    int nc = lane & 15;
    #pragma unroll
    for (int t = 0; t < 4; ++t) {
        #pragma unroll
        for (int i = 0; i < 8; ++i) {
            int m = row0 + mo + i;
            if (m < N) Y[(size_t)m * H + t * 16 + nc] = acc[t][i];
        }
    }
}

// ---------------------------------------------------------------------------
// Edge pass 1: e = leaky_relu(asrc[src]+adst[dst]); segmax[dst] = max(e)
// ---------------------------------------------------------------------------
__global__ void edge_logits_kernel(const int* __restrict__ src,
                                   const int* __restrict__ dst,
                                   const float* __restrict__ asrc,
                                   const float* __restrict__ adst,
                                   float* __restrict__ e,
                                   unsigned* __restrict__ segmax, int E) {
    int i = blockIdx.x * blockDim.x + threadIdx.x;
    if (i >= E) return;
    int s = src[i], d = dst[i];
    float v = asrc[s] + adst[d];
    v = (v > 0.0f) ? v : SLOPE * v;
    e[i] = v;
    atomicMax(&segmax[d], fenc(v));
}

// ---------------------------------------------------------------------------
// Edge pass 2: ex = exp(e - max[dst]); den[dst] += ex  (ex overwrites e)
// ---------------------------------------------------------------------------
__global__ void edge_exp_kernel(const int* __restrict__ dst,
                                const unsigned* __restrict__ segmax,
                                float* __restrict__ e,
                                float* __restrict__ den, int E) {
    int i = blockIdx.x * blockDim.x + threadIdx.x;
    if (i >= E) return;
    int d = dst[i];
    float ex = __expf(e[i] - fdec(segmax[d]));
    e[i] = ex;
    atomicAdd(&den[d], ex);
}

// ---------------------------------------------------------------------------
// Edge pass 3: out[dst] += (ex/den[dst]) * hs[src]
// 16 threads per edge, 4 features each (float4 gather + 4 atomic f32 adds)
// ---------------------------------------------------------------------------
__global__ void edge_scatter_kernel(const int* __restrict__ src,
                                    const int* __restrict__ dst,
                                    const float* __restrict__ e,
                                    const float* __restrict__ den,
                                    const float* __restrict__ hs,
                                    float* __restrict__ out, int E) {
    long tid = (long)blockIdx.x * blockDim.x + threadIdx.x;
    int i  = (int)(tid >> 4);
    int hq = ((int)tid & 15) * 4;
    if (i >= E) return;
    int s = src[i], d = dst[i];
    float alpha = e[i] / den[d];
    float4 v = *(const float4*)(hs + (size_t)s * H + hq);
    float* o = out + (size_t)d * H + hq;
    atomicAdd(o + 0, alpha * v.x);
    atomicAdd(o + 1, alpha * v.y);
    atomicAdd(o + 2, alpha * v.z);
    atomicAdd(o + 3, alpha * v.w);
}

// ---------------------------------------------------------------------------
// xi = tanh((o_txn + o_dep)*0.5) with per-relation biases folded in
// ---------------------------------------------------------------------------
__global__ void finish_int_kernel(const float* __restrict__ acc,
                                  const float* __restrict__ b0,
                                  const float* __restrict__ b2,
                                  float* __restrict__ xi, int total) {
    int idx = blockIdx.x * blockDim.x + threadIdx.x;
    if (idx >= total) return;
    int h = idx & (H - 1);
    xi[idx] = tanhf((acc[idx] + b0[h] + b2[h]) * 0.5f);
}
__global__ void finish_ext_kernel(const float* __restrict__ acc,
                                  const float* __restrict__ b1,
                                  float* __restrict__ xe, int total) {
    int idx = blockIdx.x * blockDim.x + threadIdx.x;
    if (idx >= total) return;
    int h = idx & (H - 1);
    xe[idx] = tanhf(acc[idx] + b1[h]);
}

// ---------------------------------------------------------------------------
// Final classifier: softmax(xi @ lin_W + lin_b) over C=2
// ---------------------------------------------------------------------------
__global__ void classify_kernel(const float* __restrict__ xi,
                                const float* __restrict__ linW,
                                const float* __restrict__ linb,
                                float* __restrict__ out, int N) {
    int n = blockIdx.x * blockDim.x + threadIdx.x;
    if (n >= N) return;
    const float* x = xi + (size_t)n * H;
    float z0 = linb[0], z1 = linb[1];
    #pragma unroll
    for (int d = 0; d < H; ++d) {
        float v = x[d];
        z0 += v * linW[d * C + 0];
        z1 += v * linW[d * C + 1];
    }
    float m  = fmaxf(z0, z1);
    float e0 = __expf(z0 - m), e1 = __expf(z1 - m);
    float inv = 1.0f / (e0 + e1);
    out[(size_t)n * C + 0] = e0 * inv;
    out[(size_t)n * C + 1] = e1 * inv;
}

// ---------------------------------------------------------------------------
// Host-side per-relation driver
// ---------------------------------------------------------------------------
static void run_gat(const float* xs, int Ns, const float* xd, int Nd,
                    const int* ei, int E,
                    const float* Ws, const float* Wd,
                    const float* att_s, const float* att_d,
                    float* hs, float* asrc, float* adst,
                    float* den, unsigned* segmax, float* ebuf, float* wv2,
                    float* outacc, hipStream_t stream) {
    const int* src = ei;
    const int* dst = ei + E;
    matvec64_kernel<<<1, 64, 0, stream>>>(Ws, att_s, wv2);
    matvec64_kernel<<<1, 64, 0, stream>>>(Wd, att_d, wv2 + 64);
    gemm_wmma_kernel<<<(Ns + 127) / 128, 256, 0, stream>>>(xs, Ws, hs, Ns);
    node_dot_kernel<<<(Ns + 255) / 256, 256, 0, stream>>>(xs, wv2, asrc, Ns);
    node_dot_kernel<<<(Nd + 255) / 256, 256, 0, stream>>>(xd, wv2 + 64, adst, Nd);
    zero_u32_kernel<<<(Nd + 255) / 256, 256, 0, stream>>>(segmax, Nd);
    zero_f32_kernel<<<(Nd + 255) / 256, 256, 0, stream>>>(den, Nd);
    edge_logits_kernel<<<(E + 255) / 256, 256, 0, stream>>>(src, dst, asrc, adst, ebuf, segmax, E);
    edge_exp_kernel<<<(E + 255) / 256, 256, 0, stream>>>(dst, segmax, ebuf, den, E);
    long t = (long)E * 16;
    edge_scatter_kernel<<<(int)((t + 255) / 256), 256, 0, stream>>>(src, dst, ebuf, den, hs, outacc, E);
}

extern "C" void kernel_launch(void* const* d_in, const int* in_sizes, int n_in,
                              void* d_out, int out_size, void* d_ws, size_t ws_size,
                              hipStream_t stream) {
    const float* x_int  = (const float*)d_in[0];
    const float* x_ext  = (const float*)d_in[1];
    const int*   ei_txn = (const int*)d_in[2];
    const int*   ei_wd  = (const int*)d_in[3];
    const int*   ei_dep = (const int*)d_in[4];
    const float* W_src  = (const float*)d_in[5];   // [L,3,64,64]
    const float* W_dst  = (const float*)d_in[6];   // [L,3,64,64]
    const float* att_s  = (const float*)d_in[7];   // [L,3,64]
    const float* att_d  = (const float*)d_in[8];   // [L,3,64]
    const float* bias   = (const float*)d_in[9];   // [L,3,64]
    const float* lin_W  = (const float*)d_in[10];  // [64,2]
    const float* lin_b  = (const float*)d_in[11];  // [2]

    // ---- workspace carve-up (floats) ----
    size_t off = 0;
    float* base = (float*)d_ws;
    auto take = [&](size_t nf) { float* p = base + off; off += nf; return p; };
    float*    xi_buf  = take((size_t)N_INT * H);
    float*    xe_buf  = take((size_t)N_EXT * H);
    float*    out_int = take((size_t)N_INT * H);
    float*    out_ext = take((size_t)N_EXT * H);
    float*    hs_buf  = take((size_t)N_INT * H);
    float*    asrc    = take(N_INT);
    float*    adst    = take(N_INT);
    float*    den     = take(N_INT);
    unsigned* segmax  = (unsigned*)take(N_INT);
    float*    wv2     = take(128);
    float*    ebuf    = take(E_TXN);

    auto Wmat = [&](const float* Wb, int l, int rel) { return Wb + (size_t)(l * 3 + rel) * D * H; };
    auto Avec = [&](const float* Ab, int l, int rel) { return Ab + (size_t)(l * 3 + rel) * H; };

    const float* xi = x_int;
    const float* xe = x_ext;

    for (int l = 0; l < 2; ++l) {
        zero_f32_kernel<<<(N_INT * H + 255) / 256, 256, 0, stream>>>(out_int, N_INT * H);
        zero_f32_kernel<<<(N_EXT * H + 255) / 256, 256, 0, stream>>>(out_ext, N_EXT * H);

        // rel 0: txn  (internal -> internal)
        run_gat(xi, N_INT, xi, N_INT, ei_txn, E_TXN,
                Wmat(W_src, l, 0), Wmat(W_dst, l, 0), Avec(att_s, l, 0), Avec(att_d, l, 0),
                hs_buf, asrc, adst, den, segmax, ebuf, wv2, out_int, stream);
        // rel 2: deposit (external -> internal), accumulates into same out_int
        run_gat(xe, N_EXT, xi, N_INT, ei_dep, E_DEP,
                Wmat(W_src, l, 2), Wmat(W_dst, l, 2), Avec(att_s, l, 2), Avec(att_d, l, 2),
                hs_buf, asrc, adst, den, segmax, ebuf, wv2, out_int, stream);
        // rel 1: withdraw (internal -> external)
        run_gat(xi, N_INT, xe, N_EXT, ei_wd, E_WD,
                Wmat(W_src, l, 1), Wmat(W_dst, l, 1), Avec(att_s, l, 1), Avec(att_d, l, 1),
                hs_buf, asrc, adst, den, segmax, ebuf, wv2, out_ext, stream);

        finish_int_kernel<<<(N_INT * H + 255) / 256, 256, 0, stream>>>(
            out_int, Avec(bias, l, 0), Avec(bias, l, 2), xi_buf, N_INT * H);
        finish_ext_kernel<<<(N_EXT * H + 255) / 256, 256, 0, stream>>>(
            out_ext, Avec(bias, l, 1), xe_buf, N_EXT * H);

        xi = xi_buf;
        xe = xe_buf;
    }

    classify_kernel<<<(N_INT + 255) / 256, 256, 0, stream>>>(
        xi, lin_W, lin_b, (float*)d_out, N_INT);
}